// YoloLoss_53996328845707
// MI455X (gfx1250) — compile-verified
//
#include <hip/hip_runtime.h>
#include <stdint.h>

// ---------------- CDNA5 feature detection -------------------------------
#if defined(__HIP_DEVICE_COMPILE__)
# if __has_builtin(__builtin_amdgcn_global_load_async_to_lds_b128) && \
     __has_builtin(__builtin_amdgcn_global_load_async_to_lds_b32)
#  define HAVE_ASYNC 1
# else
#  define HAVE_ASYNC 0
#  warning "gfx1250 async-to-LDS builtins not found; using sync LDS staging fallback"
# endif
#else
# define HAVE_ASYNC 0
#endif

typedef __attribute__((ext_vector_type(2))) float v2f;
typedef __attribute__((ext_vector_type(8))) float v8f;
typedef int v4i __attribute__((vector_size(4 * sizeof(int))));

#if HAVE_ASYNC
// Builtin signatures (from clang diagnostics):
//   b128: (v4i __device__*, v4i __shared__*, imm offset, imm cpol)
//   b32 : (int __device__*, int __shared__*, imm offset, imm cpol)
// HIP clang implicitly converts generic pointers for builtin args.
__device__ __forceinline__ void async_b128(const void* g, void* l) {
  __builtin_amdgcn_global_load_async_to_lds_b128((v4i*)g, (v4i*)l, 0, 0);
}
__device__ __forceinline__ void async_b32(const void* g, void* l) {
  __builtin_amdgcn_global_load_async_to_lds_b32((int*)g, (int*)l, 0, 0);
}
__device__ __forceinline__ void wait_async0() {
# if __has_builtin(__builtin_amdgcn_s_wait_asynccnt)
  __builtin_amdgcn_s_wait_asynccnt(0);
# else
  asm volatile("s_wait_asynccnt 0" ::: "memory");
# endif
}
#endif

// Stage n 4-byte elements from global to LDS, 256-thread cooperative.
// Bases are 16B aligned (block chunk starts are multiples of 256 records).
template <typename T>
__device__ __forceinline__ void stage(const T* __restrict__ g, T* l, int n, int t) {
  static_assert(sizeof(T) == 4, "4-byte elements only");
#if HAVE_ASYNC
  const int n4 = n >> 2;                       // 16-byte units
  for (int u = t; u < n4; u += 256)
    async_b128(g + u * 4, l + u * 4);
  const int off = n4 << 2;
  if (t < (n & 3))
    async_b32(g + off + t, l + off + t);
#else
  for (int u = t; u < n; u += 256) l[u] = g[u];
#endif
}

// ---------------- reductions --------------------------------------------
__device__ __forceinline__ void waveReduce4(float& a, float& b, float& c, float& d) {
  for (int m = 16; m > 0; m >>= 1) {          // wave32
    a += __shfl_xor(a, m, 32);
    b += __shfl_xor(b, m, 32);
    c += __shfl_xor(c, m, 32);
    d += __shfl_xor(d, m, 32);
  }
}

// Block-wide (256 threads = 8 waves) 4-component sum.
// Final 8x4 cross-wave combine uses V_WMMA_F32_16X16X4_F32 with an all-ones
// A matrix: D[m][n] = sum_k B[k][n]. Result: lanes 0..3 of every wave hold
// totals for components 0..3 in D[0].
__device__ __forceinline__ v8f blockReduce4(float a, float b, float c, float d,
                                            float* s_wsum /*32 floats*/) {
  waveReduce4(a, b, c, d);
  const int wave = threadIdx.x >> 5;
  const int lane = threadIdx.x & 31;
  if (lane == 0) {
    s_wsum[wave * 4 + 0] = a;
    s_wsum[wave * 4 + 1] = b;
    s_wsum[wave * 4 + 2] = c;
    s_wsum[wave * 4 + 3] = d;
  }
  __syncthreads();
  // Build B (4x16): column n = component (cols >= 4 are zero),
  // B[k][n] = wsum[k][n] + wsum[k+4][n]  -> sum over 4 rows gives 8-wave total.
  const int col = lane & 15;
  const int r0  = (lane < 16) ? 0 : 2;
  float b0 = 0.0f, b1 = 0.0f;
  if (col < 4) {
    b0 = s_wsum[r0 * 4 + col] + s_wsum[(r0 + 4) * 4 + col];
    b1 = s_wsum[(r0 + 1) * 4 + col] + s_wsum[(r0 + 5) * 4 + col];
  }
  v2f A; A.x = 1.0f; A.y = 1.0f;              // all-ones 16x4
  v2f B; B.x = b0;  B.y = b1;
  v8f C = {};
  // executed by all 8 waves with full EXEC (WMMA requirement)
  return __builtin_amdgcn_wmma_f32_16x16x4_f32(false, A, false, B,
                                               (short)0, C, false, false);
}

// ---------------- YOLO math ---------------------------------------------
__device__ __forceinline__ float iou_xyxy(float px1, float py1, float px2, float py2,
                                          float tx1, float ty1, float tx2, float ty2) {
  const float lx = fmaxf(px1, tx1), ly = fmaxf(py1, ty1);
  const float rx = fminf(px2, tx2), ry = fminf(py2, ty2);
  const float w = fmaxf(rx - lx, 0.0f), h = fmaxf(ry - ly, 0.0f);
  const float inter = w * h;
  const float ap = (px2 - px1) * (py2 - py1);   // as reference: corner diffs
  const float at = (tx2 - tx1) * (ty2 - ty1);
  return inter / (ap + at - inter);
}

// comp0 = cls_sum, comp1 = noobj_sum, comp2 = reg_sum, comp3 = contain_sum
__global__ __launch_bounds__(256) void yolo_partial(
    const float* __restrict__ pred,   // cells*30
    const float* __restrict__ tbox,   // cells*4
    const float* __restrict__ tcls,   // cells*20
    const int*   __restrict__ hmap,   // cells
    float* __restrict__ ws,           // gridDim*4 partials
    int cells) {
  __shared__ alignas(16) float s_pred[256 * 30];
  __shared__ alignas(16) float s_cls[256 * 20];
  __shared__ alignas(16) float s_box[256 * 4];
  __shared__ alignas(16) int   s_msk[256];
  __shared__ float s_wsum[32];

  const int t = threadIdx.x;
  const int base = blockIdx.x * 256;
  const int cnt = min(256, cells - base);

  stage(pred + (size_t)base * 30, s_pred, cnt * 30, t);
  stage(tcls + (size_t)base * 20, s_cls, cnt * 20, t);
  stage(tbox + (size_t)base * 4, s_box, cnt * 4, t);
  stage(hmap + (size_t)base, s_msk, cnt, t);
#if HAVE_ASYNC
  wait_async0();                 // per-wave async completion
#endif
  __syncthreads();               // cross-wave visibility

  float cls_s = 0.0f, noobj_s = 0.0f, reg_s = 0.0f, con_s = 0.0f;
  if (t < cnt) {
    const float* p  = s_pred + t * 30;
    const float* tc = s_cls + t * 20;
    const float* tb = s_box + t * 4;
    const bool obj = (s_msk[t] != 0);

    if (obj) {
      // classification: sum_j (pred_cls - target_cls)^2
      float acc = 0.0f;
#pragma unroll
      for (int j = 0; j < 20; ++j) {
        const float dd = p[10 + j] - tc[j];
        acc += dd * dd;
      }
      cls_s = acc;

      // IoU of both predicted boxes vs target
      const float tx = tb[0], ty = tb[1], tw = tb[2], th_ = tb[3];
      const float tcx = tx / 14.0f, tcy = ty / 14.0f;
      const float tx1 = tcx - 0.5f * tw, ty1 = tcy - 0.5f * th_;
      const float tx2 = tcx + 0.5f * tw, ty2 = tcy + 0.5f * th_;

      float ious[2], bx[2], by[2], bw[2], bh[2], bc[2];
#pragma unroll
      for (int b2 = 0; b2 < 2; ++b2) {
        const float x = p[b2 * 5 + 0], y = p[b2 * 5 + 1];
        const float w = p[b2 * 5 + 2], hh = p[b2 * 5 + 3];
        bx[b2] = x; by[b2] = y; bw[b2] = w; bh[b2] = hh;
        bc[b2] = p[b2 * 5 + 4];
        const float cx = x / 14.0f, cy = y / 14.0f;
        ious[b2] = iou_xyxy(cx - 0.5f * w, cy - 0.5f * hh,
                            cx + 0.5f * w, cy + 0.5f * hh,
                            tx1, ty1, tx2, ty2);
      }
      const int bi = (ious[1] > ious[0]) ? 1 : 0;   // first-max tie rule
      const float biou = fmaxf(ious[0], ious[1]);

      const float dx = bx[bi] - tx, dy = by[bi] - ty;
      const float sw = sqrtf(bw[bi]) - sqrtf(tw);
      const float sh = sqrtf(bh[bi]) - sqrtf(th_);
      reg_s = dx * dx + dy * dy + sw * sw + sh * sh;
      const float dc = bc[bi] - biou;
      con_s = dc * dc;
    } else {
      // no-object: conf^2 of both boxes
      noobj_s = p[4] * p[4] + p[9] * p[9];
    }
  }

  const v8f D = blockReduce4(cls_s, noobj_s, reg_s, con_s, s_wsum);
  if (threadIdx.x < 4)                         // wave-0 lanes 0..3, D[0][col]
    ws[blockIdx.x * 4 + threadIdx.x] = D[0];
}

__global__ __launch_bounds__(256) void yolo_final(
    const float* __restrict__ ws, int nPart, float* __restrict__ out, float invN) {
  __shared__ float s_wsum[32];
  float a = 0.0f, b = 0.0f, c = 0.0f, d = 0.0f;
  for (int i = threadIdx.x; i < nPart; i += 256) {
    a += ws[i * 4 + 0];
    b += ws[i * 4 + 1];
    c += ws[i * 4 + 2];
    d += ws[i * 4 + 3];
  }
  const v8f D = blockReduce4(a, b, c, d, s_wsum);
  const float raw = D[0];
  const int col = threadIdx.x & 3;
  // comp scaling: cls *1, noobj *L_NOOBJ(0.5), reg *L_COORD(5), contain *1 ; all /N
  const float coef = (col == 1) ? 0.5f : ((col == 2) ? 5.0f : 1.0f);
  const float scaled = raw * coef * invN;      // lanes with (lane&15)>=4: raw==0
  float tot = scaled;
  tot += __shfl_xor(tot, 1, 32);
  tot += __shfl_xor(tot, 2, 32);               // lanes 0..3 now hold the total
  if (threadIdx.x == 0) { out[0] = tot; out[4] = scaled; }  // total, cls
  else if (threadIdx.x == 1) out[3] = scaled;               // no_obj
  else if (threadIdx.x == 2) out[1] = scaled;               // reg
  else if (threadIdx.x == 3) out[2] = scaled;               // contain
}

extern "C" void kernel_launch(void* const* d_in, const int* in_sizes, int n_in,
                              void* d_out, int out_size, void* d_ws, size_t ws_size,
                              hipStream_t stream) {
  (void)n_in; (void)out_size; (void)ws_size;
  const float* pred = (const float*)d_in[0];
  const float* tbox = (const float*)d_in[1];
  const float* tcls = (const float*)d_in[2];
  const int*   hmap = (const int*)d_in[3];
  const int cells = in_sizes[3];               // N*S*S
  const int nBlk = (cells + 255) / 256;
  const float invN = 196.0f / (float)cells;    // 1/N (S*S/cells)
  float* ws = (float*)d_ws;

  yolo_partial<<<nBlk, 256, 0, stream>>>(pred, tbox, tcls, hmap, ws, cells);
  yolo_final<<<1, 256, 0, stream>>>(ws, nBlk, (float*)d_out, invN);
}